// DFAGraphEncoder_7876970020899
// MI455X (gfx1250) — compile-verified
//
#include <hip/hip_runtime.h>

typedef __attribute__((ext_vector_type(2))) float v2f;
typedef __attribute__((ext_vector_type(8))) float v8f;

#define KDIM 128   // inner dim of both GEMMs (IN_CH == HID_CH == 128)

// ---------------- degree / normalization ----------------
__global__ void k_init_deg(float* __restrict__ deg, int n) {
    int i = blockIdx.x * blockDim.x + threadIdx.x;
    if (i < n) deg[i] = 1.0f;                       // self-loop contributes 1
}

__global__ void k_count_deg(const int* __restrict__ dst, float* __restrict__ deg, int e) {
    int i = blockIdx.x * blockDim.x + threadIdx.x;
    if (i < e) atomicAdd(&deg[dst[i]], 1.0f);
}

__global__ void k_finalize_dinv(float* __restrict__ deg, int n) {
    int i = blockIdx.x * blockDim.x + threadIdx.x;
    if (i < n) deg[i] = rsqrtf(deg[i]);             // deg >= 1 always
}

// ---------------- fp32 WMMA GEMM:  H[n, OUTCH] = (relu?)(A[n, 128]) @ W[128, OUTCH] ----------------
// Block: 256 threads = 8 waves. MT m-tiles of 16 rows per block; each wave owns one
// (m-tile, n-tile) 16x16 output tile. K-loop = 32 steps of V_WMMA_F32_16X16X4_F32.
template <int OUTCH, int MT, bool RELU>
__global__ __launch_bounds__(256) void k_gemm(const float* __restrict__ A,
                                              const float* __restrict__ W,
                                              float* __restrict__ H, int n) {
    constexpr int LDSW = KDIM + 4;                  // pad: bank-conflict-free column reads
    __shared__ float tile[MT * 16 * LDSW];

    const int tid = threadIdx.x;
    const int m_base = blockIdx.x * (MT * 16);

    // cooperative stage of A tile (MT*16 rows x 128) into LDS, optional fused ReLU
    constexpr int CHUNKS = MT * 16 * (KDIM / 4);    // float4 chunks
    for (int c = tid; c < CHUNKS; c += 256) {
        int row = c >> 5;                           // 32 float4-chunks per row
        int kc  = c & 31;
        int grow = m_base + row;
        if (grow >= n) grow = n - 1;                // clamp (dup loads OK)
        float4 v = ((const float4*)(A + (size_t)grow * KDIM))[kc];
        if (RELU) {
            v.x = fmaxf(v.x, 0.0f); v.y = fmaxf(v.y, 0.0f);
            v.z = fmaxf(v.z, 0.0f); v.w = fmaxf(v.w, 0.0f);
        }
        *(float4*)&tile[row * LDSW + kc * 4] = v;
    }
    __syncthreads();

    const int wave = tid >> 5;
    const int lane = tid & 31;
    int mt, nt;
    if (OUTCH == 128) { mt = 0;         nt = wave;     }   // 1 m-tile x 8 n-tiles
    else              { mt = wave >> 2; nt = wave & 3; }   // 2 m-tiles x 4 n-tiles
    const int lm = lane & 15;           // M index (A) / N index (B,C)
    const int kh = lane >> 4;           // K-half select per ISA 16x4 layout

    const float* arow = &tile[(mt * 16 + lm) * LDSW];
    const float* wcol = W + nt * 16 + lm;

    v8f acc = {};
#pragma unroll
    for (int k0 = 0; k0 < KDIM; k0 += 4) {
        v2f a, b;
        // A 16x4 fp32 layout: lanes 0-15 hold K = k0+{0,1}, lanes 16-31 hold K = k0+{2,3}
        a.x = arow[k0 + 2 * kh + 0];
        a.y = arow[k0 + 2 * kh + 1];
        // B 4x16 fp32 layout: same K split, N striped across lanes 0-15 / 16-31
        b.x = wcol[(size_t)(k0 + 2 * kh + 0) * OUTCH];
        b.y = wcol[(size_t)(k0 + 2 * kh + 1) * OUTCH];
        acc = __builtin_amdgcn_wmma_f32_16x16x4_f32(
            /*neg_a=*/false, a, /*neg_b=*/false, b,
            /*c_mod=*/(short)0, acc, /*reuse_a=*/false, /*reuse_b=*/false);
    }

    // C/D layout: VGPR r -> M = r + 8*(lane/16), N = lane%16
    const int mrow0 = m_base + mt * 16 + (kh ? 8 : 0);
#pragma unroll
    for (int r = 0; r < 8; ++r) {
        int grow = mrow0 + r;
        if (grow < n)
            H[(size_t)grow * OUTCH + nt * 16 + lm] = acc[r];
    }
}

// ---------------- out = bias + h * dinv^2 (self-loop term), vectorized ----------------
template <int CH>
__global__ void k_seed(float* __restrict__ out, const float* __restrict__ h,
                       const float* __restrict__ bias, const float* __restrict__ dinv, int n) {
    int i = blockIdx.x * blockDim.x + threadIdx.x;
    int total = n * (CH / 4);
    if (i >= total) return;
    int node = i / (CH / 4);
    int c4   = i % (CH / 4);
    float d = dinv[node];
    float s = d * d;
    float4 hv = ((const float4*)(h + (size_t)node * CH))[c4];
    float4 bv = ((const float4*)bias)[c4];
    float4 o;
    o.x = fmaf(hv.x, s, bv.x); o.y = fmaf(hv.y, s, bv.y);
    o.z = fmaf(hv.z, s, bv.z); o.w = fmaf(hv.w, s, bv.w);
    ((float4*)(out + (size_t)node * CH))[c4] = o;
}

// ---------------- edge scatter: out[dst] += h[src] * dinv[src]*dinv[dst] ----------------
// one wave (32 lanes) per edge; lanes cover channels with float4/float2 vectors
template <int CH>
__global__ void k_scatter(const int* __restrict__ src, const int* __restrict__ dst,
                          const float* __restrict__ h, float* __restrict__ out,
                          const float* __restrict__ dinv, int e) {
    int g = blockIdx.x * blockDim.x + threadIdx.x;
    int edge = g >> 5;
    int lane = g & 31;
    if (edge >= e) return;
    int s = src[edge], d = dst[edge];
    float w = dinv[s] * dinv[d];
    if (CH == 128) {
        float4 hv = ((const float4*)(h + (size_t)s * CH))[lane];
        float* o = out + (size_t)d * CH + lane * 4;
        atomicAdd(o + 0, hv.x * w); atomicAdd(o + 1, hv.y * w);
        atomicAdd(o + 2, hv.z * w); atomicAdd(o + 3, hv.w * w);
    } else {
        float2 hv = ((const float2*)(h + (size_t)s * CH))[lane];
        float* o = out + (size_t)d * CH + lane * 2;
        atomicAdd(o + 0, hv.x * w); atomicAdd(o + 1, hv.y * w);
    }
}

extern "C" void kernel_launch(void* const* d_in, const int* in_sizes, int n_in,
                              void* d_out, int out_size, void* d_ws, size_t ws_size,
                              hipStream_t stream) {
    const float* x  = (const float*)d_in[0];
    const int*   ei = (const int*)d_in[1];
    const float* W1 = (const float*)d_in[2];
    const float* b1 = (const float*)d_in[3];
    const float* W2 = (const float*)d_in[4];
    const float* b2 = (const float*)d_in[5];
    float* out = (float*)d_out;

    const int n = in_sizes[0] / 128;     // 100000 nodes
    const int e = in_sizes[1] / 2;       // 1.6M edges
    const int* src = ei;                 // edge_index[0]
    const int* dst = ei + e;             // edge_index[1]

    // workspace: dinv[n] | buf h[n*128] (h1, then reused as h2) | buf o1[n*128]
    float* dinv = (float*)d_ws;
    float* h  = dinv + n;
    float* o1 = h + (size_t)n * 128;

    const unsigned T = 256;
    // normalization
    k_init_deg<<<(n + T - 1) / T, T, 0, stream>>>(dinv, n);
    k_count_deg<<<(e + T - 1) / T, T, 0, stream>>>(dst, dinv, e);
    k_finalize_dinv<<<(n + T - 1) / T, T, 0, stream>>>(dinv, n);

    // layer 1: h1 = x @ W1 ; o1 = b1 + self-loop + edge scatter
    k_gemm<128, 1, false><<<(n + 15) / 16, T, 0, stream>>>(x, W1, h, n);
    k_seed<128><<<((unsigned)(n * 32) + T - 1) / T, T, 0, stream>>>(o1, h, b1, dinv, n);
    k_scatter<128><<<(unsigned)((e + 7) / 8), T, 0, stream>>>(src, dst, h, o1, dinv, e);

    // layer 2: h2 = relu(o1) @ W2 (ReLU fused into A-tile load); out = b2 + self-loop + scatter
    k_gemm<64, 2, true><<<(n + 31) / 32, T, 0, stream>>>(o1, W2, h, n);
    k_seed<64><<<((unsigned)(n * 16) + T - 1) / T, T, 0, stream>>>(out, h, b2, dinv, n);
    k_scatter<64><<<(unsigned)((e + 7) / 8), T, 0, stream>>>(src, dst, h, out, dinv, e);
}